// TMMSpectrum_6897717477750
// MI455X (gfx1250) — compile-verified
//
#include <hip/hip_runtime.h>
#include <math.h>

#define Bsz 64
#define Ssz 32
#define Vsz 512
#define Wsz 2048

typedef float v2f __attribute__((ext_vector_type(2)));
typedef float v8f __attribute__((ext_vector_type(8)));

// ---------------- complex double helpers ----------------
struct Cd { double re, im; };
__device__ __forceinline__ Cd cadd(Cd a, Cd b){ return {a.re+b.re, a.im+b.im}; }
__device__ __forceinline__ Cd csub(Cd a, Cd b){ return {a.re-b.re, a.im-b.im}; }
__device__ __forceinline__ Cd cmul(Cd a, Cd b){ return {a.re*b.re - a.im*b.im, a.re*b.im + a.im*b.re}; }
__device__ __forceinline__ Cd cscale(Cd a, double s){ return {a.re*s, a.im*s}; }
__device__ __forceinline__ Cd cdiv(Cd a, Cd b){
  double id = 1.0/(b.re*b.re + b.im*b.im);
  return {(a.re*b.re + a.im*b.im)*id, (a.im*b.re - a.re*b.im)*id};
}
__device__ __forceinline__ Cd cinv(Cd b){
  double id = 1.0/(b.re*b.re + b.im*b.im);
  return { b.re*id, -b.im*id};
}
__device__ __forceinline__ Cd csqrt_(Cd z){
  double m  = sqrt(z.re*z.re + z.im*z.im);
  double re = sqrt(fmax(0.5*(m + z.re), 0.0));
  double im = copysign(sqrt(fmax(0.5*(m - z.re), 0.0)), z.im);
  return {re, im};
}
// f = n * cos(arcsin(s0/n)) = n * sqrt(1 - (s0/n)^2)  (principal branches)
__device__ __forceinline__ Cd layer_f(Cd n, double s0){
  Cd w  = cdiv({s0, 0.0}, n);
  Cd w2 = cmul(w, w);
  Cd ct = csqrt_({1.0 - w2.re, -w2.im});
  return cmul(n, ct);
}
__device__ __forceinline__ float clean01(float x){
  if (!isfinite(x)) x = 0.f;            // nan/inf -> 0 (matches nan_to_num)
  return fminf(fmaxf(x, 0.f), 1.f);
}

// ---------------- kernel 1: gather nk_per_token (f64 -> f32) ----------------
__global__ __launch_bounds__(256) void gather_nk(const double* __restrict__ nkr,
                                                 const double* __restrict__ nki,
                                                 const int* __restrict__ mat_idx,
                                                 float* __restrict__ tr,
                                                 float* __restrict__ ti){
  int i = blockIdx.x * 256 + threadIdx.x;       // 0 .. V*W-1
  int v = i / Wsz, w = i % Wsz;
  int m = mat_idx[v];
  tr[i] = (float)nkr[m*Wsz + w];
  ti[i] = (float)nki[m*Wsz + w];
}

// ---------------- kernel 2: row normalize + thickness dot ----------------
__global__ __launch_bounds__(256) void prep(const float* __restrict__ stacks,
                                            const double* __restrict__ thick,
                                            const int* eosp, const int* padp, const int* mskp,
                                            float* __restrict__ p_norm,
                                            double* __restrict__ t_base,
                                            int* __restrict__ is_zero){
  __shared__ float  sred[256];
  __shared__ double dred[256];
  int r   = blockIdx.x;           // 0 .. B*S-1
  int tid = threadIdx.x;
  int eos = eosp[0], pad = padp[0], msk = mskp[0];
  int v0 = tid*2, v1 = tid*2 + 1;
  float x0 = stacks[r*Vsz + v0]; if (v0==eos || v0==pad || v0==msk) x0 = 0.f;
  float x1 = stacks[r*Vsz + v1]; if (v1==eos || v1==pad || v1==msk) x1 = 0.f;
  sred[tid] = x0 + x1;
  __syncthreads();
  for (int s = 128; s > 0; s >>= 1){ if (tid < s) sred[tid] += sred[tid+s]; __syncthreads(); }
  float Z  = sred[0];
  bool  zr = !(Z > 0.f);
  float inv = zr ? 1.f : 1.f / fmaxf(Z, 1e-8f);
  float p0 = x0 * inv, p1 = x1 * inv;
  p_norm[r*Vsz + v0] = p0;
  p_norm[r*Vsz + v1] = p1;
  dred[tid] = (double)p0 * thick[v0] + (double)p1 * thick[v1];
  __syncthreads();
  for (int s = 128; s > 0; s >>= 1){ if (tid < s) dred[tid] += dred[tid+s]; __syncthreads(); }
  if (tid == 0){ t_base[r] = zr ? 0.0 : dred[0]; is_zero[r] = zr ? 1 : 0; }
}

// ---------------- kernel 3: survival cumprod applied to t_base ----------------
__global__ __launch_bounds__(64) void active_k(const float* __restrict__ stacks,
                                               const int* eosp,
                                               double* __restrict__ t_base){
  int b = blockIdx.x * blockDim.x + threadIdx.x;
  if (b >= Bsz) return;
  int eos = eosp[0];
  double surv = 1.0;                               // active[0] = 1
  for (int s = 0; s < Ssz; ++s){
    t_base[b*Ssz + s] *= surv;
    float pe = fminf(fmaxf(stacks[(b*Ssz + s)*Vsz + eos], 0.f), 1.f);
    surv *= (1.0 - (double)pe + 1e-12);
  }
}

// ---------------- kernel 4: WMMA GEMM  n_base = p_norm @ nk_tok ----------------
// A: [2048 x 512] row-major, B: [512 x 2048] row-major, D: [2048 x 2048]
// One wave per 16x16 output tile; real+imag share the A fragment.
__global__ __launch_bounds__(256) void gemm_wmma(const float* __restrict__ A,
                                                 const float* __restrict__ Br,
                                                 const float* __restrict__ Bi,
                                                 float* __restrict__ Dr,
                                                 float* __restrict__ Di){
  int lane = threadIdx.x & 31;
  int wave = threadIdx.x >> 5;
  int n0 = (blockIdx.x * 8 + wave) * 16;
  int m0 = blockIdx.y * 16;
  int hl = lane >> 4;        // lane half: K offset 0 or 2
  int l  = lane & 15;
  v8f accr = {}; v8f acci = {};
  const float* arow = A + (m0 + l) * Vsz;
  #pragma unroll 4
  for (int k = 0; k < Vsz; k += 4) {
    int ka = k + hl*2;
    float2 at = *(const float2*)(arow + ka);      // A: VGPR0=K(ka), VGPR1=K(ka+1)
    v2f av; av[0] = at.x; av[1] = at.y;
    v2f br, bi;                                    // B: row-striped across lanes
    br[0] = Br[ ka      *Wsz + n0 + l];
    br[1] = Br[(ka + 1) *Wsz + n0 + l];
    bi[0] = Bi[ ka      *Wsz + n0 + l];
    bi[1] = Bi[(ka + 1) *Wsz + n0 + l];
    accr = __builtin_amdgcn_wmma_f32_16x16x4_f32(false, av, false, br, (short)0, accr, false, false);
    acci = __builtin_amdgcn_wmma_f32_16x16x4_f32(false, av, false, bi, (short)0, acci, false, false);
  }
  int col = n0 + l;
  #pragma unroll
  for (int i = 0; i < 8; ++i){
    int row = m0 + i + hl*8;                       // C/D layout: VGPRi = M=i / M=i+8
    Dr[row*Wsz + col] = accr[i];
    Di[row*Wsz + col] = acci[i];
  }
}

// ---------------- kernel 5: TMM scan per (b, w) ----------------
__global__ __launch_bounds__(256) void tmm(const float* __restrict__ nbr,
                                           const float* __restrict__ nbi,
                                           const double* __restrict__ t_fin,
                                           const int* __restrict__ is_zero,
                                           const float* __restrict__ wl,
                                           const float* __restrict__ theta,
                                           const double* __restrict__ nkr,
                                           const double* __restrict__ nki,
                                           const int* __restrict__ subp,
                                           float* __restrict__ out){
  int idx = blockIdx.x * blockDim.x + threadIdx.x;  // 0 .. B*W-1
  int b = idx / Wsz, w = idx % Wsz;
  int sub = subp[0];
  double lam  = (double)wl[w];
  double th0  = (double)theta[0];
  double s0   = sin(th0);
  double c0   = sqrt(fmax(1.0 - s0*s0, 0.0));
  Cd f0 = {c0, 0.0};

  auto load_f = [&](int s) -> Cd {
    Cd n;
    if (is_zero[b*Ssz + s]) { n.re = nkr[sub*Wsz + w]; n.im = nki[sub*Wsz + w]; }
    else { n.re = (double)nbr[(b*Ssz + s)*Wsz + w]; n.im = (double)nbi[(b*Ssz + s)*Wsz + w]; }
    return layer_f(n, s0);
  };

  // interface 0 -> 1
  Cd fk  = load_f(0);
  Cd sm  = cadd(f0, fk);
  Cd r0  = cdiv(csub(f0, fk), sm);
  Cd it0 = cinv(cdiv(cscale(f0, 2.0), sm));
  Cd Am = it0, Bm = cmul(r0, it0), Cm = Bm, Dm = Am;

  double twopi_lam = 2.0 * M_PI / lam;
  for (int k = 1; k <= Ssz; ++k){
    double t  = t_fin[b*Ssz + (k-1)];
    double dr = fk.re * twopi_lam * t;
    double di = fmin(fk.im * twopi_lam * t, 35.0);     // DELTA_IMAG_CLAMP
    double e  = exp(di), ei = exp(-di);
    double cs = cos(dr), sn = sin(dr);
    Cd P = { e*cs,  -e*sn };     // exp(-i*delta)
    Cd Q = { ei*cs,  ei*sn };    // exp(+i*delta)
    Cd fn = (k < Ssz) ? load_f(k) : f0;                // last layer n = 1
    Cd s2 = cadd(fk, fn);
    Cd rj = cdiv(csub(fk, fn), s2);
    Cd it = cinv(cdiv(cscale(fk, 2.0), s2));
    Cd aP = cmul(Am, P), bQ = cmul(Bm, Q);
    Cd cP = cmul(Cm, P), dQ = cmul(Dm, Q);
    Cd A2 = cmul(cadd(aP, cmul(bQ, rj)), it);
    Cd B2 = cmul(cadd(cmul(aP, rj), bQ), it);
    Cd C2 = cmul(cadd(cP, cmul(dQ, rj)), it);
    Cd D2 = cmul(cadd(cmul(cP, rj), dQ), it);
    Am = A2; Bm = B2; Cm = C2; Dm = D2;
    fk = fn;
  }
  Cd ramp = cdiv(Cm, Am);
  Cd tamp = cinv(Am);
  double R = ramp.re*ramp.re + ramp.im*ramp.im;
  double T = (tamp.re*tamp.re + tamp.im*tamp.im) * (fk.re / f0.re);
  float Rf = clean01((float)R);
  float Tf = clean01((float)T);
  float Af = clean01(1.f - Rf - Tf);
  out[b*3*Wsz           + w] = Rf;
  out[b*3*Wsz +   Wsz   + w] = Af;
  out[b*3*Wsz + 2*Wsz   + w] = Tf;
}

extern "C" void kernel_launch(void* const* d_in, const int* in_sizes, int n_in,
                              void* d_out, int out_size, void* d_ws, size_t ws_size,
                              hipStream_t stream) {
  const float*  stacks  = (const float*) d_in[0];
  const float*  wl      = (const float*) d_in[1];
  const float*  theta   = (const float*) d_in[2];
  const double* nkr     = (const double*)d_in[3];
  const double* nki     = (const double*)d_in[4];
  const double* thick   = (const double*)d_in[5];
  const int*    mat_idx = (const int*)   d_in[6];
  const int*    subp    = (const int*)   d_in[7];
  const int*    eosp    = (const int*)   d_in[8];
  const int*    padp    = (const int*)   d_in[9];
  const int*    mskp    = (const int*)   d_in[10];
  float* out = (float*)d_out;

  char* ws = (char*)d_ws;
  float*  p_norm  = (float*)(ws);                       //  4 MB
  float*  nk_tr   = (float*)(ws +  4194304);            //  4 MB
  float*  nk_ti   = (float*)(ws +  8388608);            //  4 MB
  float*  nbr     = (float*)(ws + 12582912);            // 16 MB
  float*  nbi     = (float*)(ws + 29360128);            // 16 MB
  double* t_base  = (double*)(ws + 46137344);           // 16 KB
  int*    is_zero = (int*)   (ws + 46153728);           //  8 KB

  hipLaunchKernelGGL(gather_nk, dim3((Vsz*Wsz)/256), dim3(256), 0, stream,
                     nkr, nki, mat_idx, nk_tr, nk_ti);
  hipLaunchKernelGGL(prep, dim3(Bsz*Ssz), dim3(256), 0, stream,
                     stacks, thick, eosp, padp, mskp, p_norm, t_base, is_zero);
  hipLaunchKernelGGL(active_k, dim3(1), dim3(64), 0, stream,
                     stacks, eosp, t_base);
  hipLaunchKernelGGL(gemm_wmma, dim3(Wsz/(16*8), (Bsz*Ssz)/16), dim3(256), 0, stream,
                     p_norm, nk_tr, nk_ti, nbr, nbi);
  hipLaunchKernelGGL(tmm, dim3((Bsz*Wsz)/256), dim3(256), 0, stream,
                     nbr, nbi, t_base, is_zero, wl, theta, nkr, nki, subp, out);
}